// RAConv_44332652430045
// MI455X (gfx1250) — compile-verified
//
#include <hip/hip_runtime.h>

#define HW     16384      // 128*128 spatial positions per batch
#define CIN    64
#define KTOT   576        // Cin * 3 * 3
#define COUT   128
#define NB     8
#define TSIZE  32768      // hash-table slots per batch (load factor <= 0.5)
#define EMPTYK (-2147483647 - 1)

typedef __attribute__((ext_vector_type(2))) float v2f;
typedef __attribute__((ext_vector_type(8))) float v8f;

// ---------------------------------------------------------------- channel sum
__global__ __launch_bounds__(256)
void csum_kernel(const float* __restrict__ fmap, float* __restrict__ csum) {
    int idx = blockIdx.x * 256 + threadIdx.x;           // over NB*HW
    if (idx >= NB * HW) return;
    int n = idx >> 14, hw = idx & (HW - 1);
    const float* p = fmap + (size_t)n * CIN * HW + hw;
    float s = 0.f;
#pragma unroll 8
    for (int c = 0; c < CIN; ++c) s += p[(size_t)c * HW];
    csum[idx] = s;
}

// --------------------------------------------- q = trunc(mean(patch) * 1000)
__global__ __launch_bounds__(256)
void qmap_kernel(const float* __restrict__ csum, int* __restrict__ q) {
    int idx = blockIdx.x * 256 + threadIdx.x;
    if (idx >= NB * HW) return;
    int n = idx >> 14, l = idx & (HW - 1);
    int h = l >> 7, w = l & 127;
    const float* cs = csum + (size_t)n * HW;
    float s = 0.f;
#pragma unroll
    for (int di = -1; di <= 1; ++di)
#pragma unroll
        for (int dj = -1; dj <= 1; ++dj) {
            int hh = h + di, ww = w + dj;
            if ((unsigned)hh < 128u && (unsigned)ww < 128u)
                s += cs[(hh << 7) + ww];
        }
    float mean = s / 576.0f;          // matches cols.mean(axis=1)
    q[idx] = (int)(mean * 1000.0f);   // C cast truncates toward zero, like astype(int32)
}

// ------------------------------------------------------------- hash utilities
__device__ __forceinline__ unsigned hash32(int x) {
    unsigned h = (unsigned)x;
    h ^= h >> 16; h *= 0x7feb352du;
    h ^= h >> 15; h *= 0x846ca68bu;
    h ^= h >> 16;
    return h;
}

__global__ __launch_bounds__(256)
void tinit_kernel(int* __restrict__ keys, int* __restrict__ vals) {
    int idx = blockIdx.x * 256 + threadIdx.x;
    if (idx >= NB * TSIZE) return;
    keys[idx] = EMPTYK;
    vals[idx] = 0x7FFFFFFF;
}

__global__ __launch_bounds__(256)
void tinsert_kernel(const int* __restrict__ q, int* __restrict__ keys, int* __restrict__ vals) {
    int idx = blockIdx.x * 256 + threadIdx.x;
    if (idx >= NB * HW) return;
    int n = idx >> 14, l = idx & (HW - 1);
    int qq = q[idx];
    int* K = keys + n * TSIZE;
    int* V = vals + n * TSIZE;
    unsigned slot = hash32(qq) & (TSIZE - 1);
    for (;;) {
        int old = atomicCAS(&K[slot], EMPTYK, qq);
        if (old == EMPTYK || old == qq) { atomicMin(&V[slot], l); break; }
        slot = (slot + 1) & (TSIZE - 1);
    }
}

__global__ __launch_bounds__(256)
void tlookup_kernel(const int* __restrict__ q, const int* __restrict__ keys,
                    const int* __restrict__ vals, int* __restrict__ rep) {
    int idx = blockIdx.x * 256 + threadIdx.x;
    if (idx >= NB * HW) return;
    int n = idx >> 14;
    int qq = q[idx];
    const int* K = keys + n * TSIZE;
    const int* V = vals + n * TSIZE;
    unsigned slot = hash32(qq) & (TSIZE - 1);
    while (K[slot] != qq) slot = (slot + 1) & (TSIZE - 1);
    rep[idx] = V[slot];
}

// ------------------------- gather-patch implicit GEMM with fp32 WMMA (16x16x4)
// block: 256 threads = 8 wave32; covers 16 positions (N) x 128 out-channels (M)
__global__ __launch_bounds__(256)
void raconv_gemm_kernel(const float* __restrict__ fmap, const float* __restrict__ Wt,
                        const float* __restrict__ bias, const int* __restrict__ rep,
                        float* __restrict__ out) {
    __shared__ float G[KTOT * 16];   // gathered patch tile, [k][p] layout, 36 KB
    __shared__ int HH[16], WW[16];

    const int n  = blockIdx.y;
    const int l0 = blockIdx.x << 4;
    const int tid = threadIdx.x;

    if (tid < 16) {
        int r = rep[n * HW + l0 + tid];   // representative position for column p
        HH[tid] = r >> 7;
        WW[tid] = r & 127;
    }
    __syncthreads();

    // stage gathered 576x16 patch tile into LDS (zero-padded 3x3 neighborhoods)
    const float* fn = fmap + (size_t)n * CIN * HW;
    for (int e = tid; e < KTOT * 16; e += 256) {
        int k = e >> 4, p = e & 15;
        int c = k / 9, kk = k - c * 9;       // channel-major: k = c*9 + (ki*3+kj)
        int ki = kk / 3, kj = kk - ki * 3;
        int h = HH[p] + ki - 1, w = WW[p] + kj - 1;
        float v = 0.f;
        if ((unsigned)h < 128u && (unsigned)w < 128u)
            v = fn[((size_t)c << 14) + (h << 7) + w];
        G[(k << 4) + p] = v;
    }
    __syncthreads();

    const int lane = tid & 31;
    const int wave = tid >> 5;         // wave w handles out-channels [w*16, w*16+16)
    const int col  = lane & 15;        // M index (A) / N index (B,D)
    const int half = lane >> 4;        // 0: K={0,1}, 1: K={2,3}
    const int koff = half << 1;

    const float* wp = Wt + (size_t)(wave * 16 + col) * KTOT + koff;  // 8B aligned
    const float* gp = &G[(koff << 4) + col];

    v8f acc = {};
#pragma unroll 4
    for (int k0 = 0; k0 < KTOT; k0 += 4) {
        v2f a = *(const v2f*)(wp + k0);        // A[M=col][k0+koff, k0+koff+1]
        v2f b;
        b.x = gp[(k0 << 4)];                   // B[k0+koff  ][N=col]
        b.y = gp[(k0 << 4) + 16];              // B[k0+koff+1][N=col]
        acc = __builtin_amdgcn_wmma_f32_16x16x4_f32(
            /*neg_a=*/false, a, /*neg_b=*/false, b,
            /*c_mod=*/(short)0, acc, /*reuse_a=*/false, /*reuse_b=*/false);
    }

    // D layout: VGPR r -> M = wave*16 + r + half*8, N = col
    float* op = out + ((size_t)(n * COUT + wave * 16 + half * 8) << 14) + l0 + col;
    const float* bp = bias + wave * 16 + half * 8;
#pragma unroll
    for (int r = 0; r < 8; ++r)
        op[(size_t)r << 14] = acc[r] + bp[r];
}

// ---------------------------------------------------------------------- launch
extern "C" void kernel_launch(void* const* d_in, const int* in_sizes, int n_in,
                              void* d_out, int out_size, void* d_ws, size_t ws_size,
                              hipStream_t stream) {
    const float* fmap   = (const float*)d_in[0];   // (8,64,128,128)
    const float* weight = (const float*)d_in[1];   // (128,576)
    const float* bias   = (const float*)d_in[2];   // (128,1)
    float* out = (float*)d_out;                    // (8,128,128,128)

    // workspace layout (all 4-byte types): ~3.7 MB total
    float* csum = (float*)d_ws;                    // NB*HW floats
    int*   q    = (int*)(csum + NB * HW);          // NB*HW ints
    int*   rep  = q + NB * HW;                     // NB*HW ints
    int*   keys = rep + NB * HW;                   // NB*TSIZE ints
    int*   vals = keys + NB * TSIZE;               // NB*TSIZE ints

    const int total = NB * HW;
    csum_kernel   <<<(total + 255) / 256, 256, 0, stream>>>(fmap, csum);
    qmap_kernel   <<<(total + 255) / 256, 256, 0, stream>>>(csum, q);
    tinit_kernel  <<<(NB * TSIZE + 255) / 256, 256, 0, stream>>>(keys, vals);
    tinsert_kernel<<<(total + 255) / 256, 256, 0, stream>>>(q, keys, vals);
    tlookup_kernel<<<(total + 255) / 256, 256, 0, stream>>>(q, keys, vals, rep);

    dim3 grid(HW / 16, NB);
    raconv_gemm_kernel<<<grid, 256, 0, stream>>>(fmap, weight, bias, rep, out);
    (void)in_sizes; (void)n_in; (void)out_size; (void)ws_size;
}